// SPMC_53317724013320
// MI455X (gfx1250) — compile-verified
//
#include <hip/hip_runtime.h>

// SPMC forward warp (scatter-splat with bilinear weights) for MI455X (gfx1250).
//
// Strategy (memory-bound op, no matmul content -> no WMMA):
//   1) spmc_zero: vectorized 128-bit zero-fill of the 176 MB output through L2
//      (regular temporal stores so the lines stay resident in the 192 MB L2).
//   2) spmc_splat: one thread per source pixel; coalesced flow/img loads,
//      4 bounds-checked corners x 3 channels of non-returning
//      global_atomic_add_f32 that hit the freshly-zeroed L2-resident output.

constexpr int kB = 8;
constexpr int kC = 3;
constexpr int kH = 256;
constexpr int kW = 448;
constexpr int kN = kH * kW;          // 114688 pixels per (b) image plane
constexpr int kTotal = kB * kN;      // 917504 source pixels

__global__ __launch_bounds__(256) void spmc_zero(float4* __restrict__ out, int n4) {
  int i = blockIdx.x * blockDim.x + threadIdx.x;
  if (i < n4) out[i] = float4{0.f, 0.f, 0.f, 0.f};   // global_store_b128, TH=RT (stay in L2)
}

__global__ __launch_bounds__(256) void spmc_zero_tail(float* __restrict__ out,
                                                      int start, int n) {
  int i = start + blockIdx.x * blockDim.x + threadIdx.x;
  if (i < n) out[i] = 0.f;
}

__global__ __launch_bounds__(256) void spmc_splat(const float* __restrict__ img,
                                                  const float* __restrict__ flow,
                                                  const int* __restrict__ scale_p,
                                                  float* __restrict__ out) {
  int tid = blockIdx.x * blockDim.x + threadIdx.x;
  if (tid >= kTotal) return;

  int b = tid / kN;
  int p = tid - b * kN;        // h*W + w within the image
  int h = p / kW;
  int w = p - h * kW;

  // Uniform scalar (1-element device array); L2-hit broadcast load.
  int scale = *scale_p;
  int Hs = kH * scale;
  int Ws = kW * scale;

  // Coalesced input reads. Channel planes are H*W apart: prefetch the far
  // planes (global_prefetch_b8) while the flow math is in flight.
  const float* imgb  = img  + (size_t)b * kC * kN + p;
  const float* flowb = flow + (size_t)b * 2  * kN + p;
  __builtin_prefetch(imgb + kN,     0, 1);
  __builtin_prefetch(imgb + 2 * kN, 0, 1);

  float fu = flowb[0];
  float fv = flowb[kN];
  float c0 = imgb[0];
  float c1 = imgb[kN];
  float c2 = imgb[2 * kN];

  float sf = (float)scale;
  float u = ((float)w + fu) * sf;
  float v = ((float)h + fv) * sf;
  float x0f = floorf(u);
  float y0f = floorf(v);
  float fx = u - x0f;
  float fy = v - y0f;
  int x0 = (int)x0f;
  int y0 = (int)y0f;

  float w00 = (1.f - fx) * (1.f - fy);
  float w10 = fx * (1.f - fy);
  float w01 = (1.f - fx) * fy;
  float w11 = fx * fy;

  size_t plane = (size_t)Hs * Ws;
  float* outb = out + (size_t)b * kC * plane;

  const int   xs[4] = {x0, x0 + 1, x0,     x0 + 1};
  const int   ys[4] = {y0, y0,     y0 + 1, y0 + 1};
  const float wk[4] = {w00, w10, w01, w11};

#pragma unroll
  for (int k = 0; k < 4; ++k) {
    int xi = xs[k];
    int yi = ys[k];
    // Reference zeroes the weight for OOB corners and adds to a clamped index;
    // adding 0.0f is a no-op, so skipping OOB corners is numerically identical.
    if ((xi >= 0) & (xi < Ws) & (yi >= 0) & (yi < Hs)) {
      size_t idx = (size_t)yi * Ws + (size_t)xi;
      float g = wk[k];
      // Non-returning global_atomic_add_f32 (cheapest atomic form); output is
      // L2-resident after the zero-fill, so these resolve in the 192 MB L2.
      atomicAdd(outb + idx,             c0 * g);
      atomicAdd(outb + plane + idx,     c1 * g);
      atomicAdd(outb + 2 * plane + idx, c2 * g);
    }
  }
}

extern "C" void kernel_launch(void* const* d_in, const int* in_sizes, int n_in,
                              void* d_out, int out_size, void* d_ws, size_t ws_size,
                              hipStream_t stream) {
  const float* img   = (const float*)d_in[0];
  const float* flow  = (const float*)d_in[1];
  const int*   scale = (const int*)d_in[2];
  float* out = (float*)d_out;

  // 1) Zero the output through L2 (keeps the 176 MB output resident for atomics).
  int n4   = out_size >> 2;
  int tail = out_size - (n4 << 2);
  if (n4 > 0) {
    int zb = (n4 + 255) / 256;
    spmc_zero<<<zb, 256, 0, stream>>>((float4*)out, n4);
  }
  if (tail > 0) {
    spmc_zero_tail<<<1, 256, 0, stream>>>(out, n4 << 2, out_size);
  }

  // 2) Scatter-splat: one thread per source pixel (stream-ordered after fill).
  int sb = (kTotal + 255) / 256;
  spmc_splat<<<sb, 256, 0, stream>>>(img, flow, scale, out);
}